// u_quant_weight_linear_2628519985351
// MI455X (gfx1250) — compile-verified
//
#include <hip/hip_runtime.h>

// Fake-quant forward: out[o,i] = clamp(rint(w[o,i]/s_o), -q_o, q_o) * s_o
// s_o = max(alh[o], 1e-5); q_o = max(2^(rint(clamp(|bit[o]|,1,6))-1)-1, 1)
//
// Memory-bound streaming kernel: 361 MB traffic -> ~15.5us floor @ 23.3 TB/s.
// b128 non-temporal loads/stores (dataset > 192MB L2), SGPR per-row scalars,
// wave32, one workgroup (8 waves) per output channel.

typedef float v4f __attribute__((ext_vector_type(4)));

#define OUT_CH     4096
#define IN_FEAT    11008
#define V4_PER_ROW (IN_FEAT / 4)   // 2752
#define THREADS    256
#define FULL_ITERS (V4_PER_ROW / THREADS)  // 10
#define EPS        1e-5f
#define BIT_BOUND  6.0f

__device__ __forceinline__ void quant_v4(const v4f* __restrict__ wrow,
                                         v4f* __restrict__ orow,
                                         int c, float inv_s, float s,
                                         float qmin, float qmax)
{
    v4f v = __builtin_nontemporal_load(&wrow[c]);   // global_load_b128, TH=NT
    v4f r;
#pragma unroll
    for (int k = 0; k < 4; ++k) {
        float q = rintf(v[k] * inv_s);              // v_rndne_f32 (round-half-even, matches jnp.round)
        q = fminf(fmaxf(q, qmin), qmax);            // v_max_num / v_min_num
        r[k] = q * s;
    }
    __builtin_nontemporal_store(r, &orow[c]);       // global_store_b128, TH=NT
}

__global__ __launch_bounds__(THREADS) void fakequant_rows_kernel(
    const float* __restrict__ w,
    const float* __restrict__ alh,
    const float* __restrict__ bit,
    float* __restrict__ out)
{
    const int row = blockIdx.x;                     // uniform -> SGPR addressing

    // Per-row scalars: hoisted divide + exp2, once per block (s_load path).
    const float s     = fmaxf(alh[row], EPS);
    const float b_eff = fminf(fmaxf(fabsf(bit[row]), 1.0f), BIT_BOUND);
    const float qmax  = fmaxf(exp2f(rintf(b_eff) - 1.0f) - 1.0f, 1.0f);
    const float qmin  = -qmax;
    const float inv_s = 1.0f / s;                   // one precise divide per block

    const v4f* __restrict__ wrow = (const v4f*)(w   + (size_t)row * IN_FEAT);
    v4f*       __restrict__ orow = (v4f*)      (out + (size_t)row * IN_FEAT);

    int c = threadIdx.x;
#pragma unroll 2
    for (int i = 0; i < FULL_ITERS; ++i, c += THREADS) {
        // Cover latency of the next tile while this one computes.
        __builtin_prefetch(&wrow[c + THREADS], 0, 0);   // global_prefetch_b8
        quant_v4(wrow, orow, c, inv_s, s, qmin, qmax);
    }
    if (c < V4_PER_ROW)                                  // tail: lanes tid < 192
        quant_v4(wrow, orow, c, inv_s, s, qmin, qmax);
}

extern "C" void kernel_launch(void* const* d_in, const int* in_sizes, int n_in,
                              void* d_out, int out_size, void* d_ws, size_t ws_size,
                              hipStream_t stream)
{
    (void)in_sizes; (void)n_in; (void)d_ws; (void)ws_size; (void)out_size;

    const float* w   = (const float*)d_in[0];   // [4096, 11008] f32
    const float* alh = (const float*)d_in[1];   // [4096, 1]     f32
    const float* bit = (const float*)d_in[2];   // [4096, 1]     f32
    // d_in[3] = init_weight_bit (unused in forward)
    float* out = (float*)d_out;                 // [4096, 11008] f32

    fakequant_rows_kernel<<<dim3(OUT_CH), dim3(THREADS), 0, stream>>>(w, alh, bit, out);
}